// LiftSplatShoot_8091718385932
// MI455X (gfx1250) — compile-verified
//
#include <hip/hip_runtime.h>
#include <hip/hip_bf16.h>
#include <cstdint>
#include <math.h>

// ---------------- problem constants (match reference) ----------------
#define LSS_B   8
#define LSS_N   6
#define LSS_D   41
#define LSS_C   64
#define LSS_FH  16
#define LSS_FW  44
#define LSS_CH  (LSS_D + LSS_C)          // 105 channels in depthnet output
#define LSS_PP  (LSS_FH * LSS_FW)        // 704 pixels per (b,n)
#define NXX 200
#define NXY 200
#define BEV_CELLS (NXX * NXY)            // 40000 voxels per batch (z==1)
#define BEV_ELEMS ((size_t)LSS_B * BEV_CELLS * LSS_C) // 20,480,000 f32

// ---------------- gfx1250 async global->LDS staging ----------------
#if __has_builtin(__builtin_amdgcn_global_load_async_to_lds_b32)
#define ASYNC_BUILTIN 1
#else
#define ASYNC_BUILTIN 0
#endif
#if __has_builtin(__builtin_amdgcn_s_wait_asynccnt)
#define WAIT_BUILTIN 1
#else
#define WAIT_BUILTIN 0
#endif

typedef __attribute__((address_space(1))) int as1_int;  // global
typedef __attribute__((address_space(3))) int as3_int;  // LDS

// One b32 per lane: global (per-lane addr) -> LDS (per-lane addr), ASYNCcnt-tracked.
__device__ __forceinline__ void async_ld32(const float* gsrc, const float* ldst) {
#if ASYNC_BUILTIN
  __builtin_amdgcn_global_load_async_to_lds_b32(
      (as1_int*)(uintptr_t)gsrc,
      (as3_int*)(uint32_t)(uintptr_t)ldst,  // low 32 bits of a generic LDS ptr = LDS offset
      0, 0);
#else
  uint64_t ga = (uint64_t)(uintptr_t)gsrc;
  uint32_t la = (uint32_t)(uintptr_t)ldst;
  asm volatile("global_load_async_to_lds_b32 %0, %1, off" :: "v"(la), "v"(ga) : "memory");
#endif
}

__device__ __forceinline__ void async_wait0() {
#if WAIT_BUILTIN
  __builtin_amdgcn_s_wait_asynccnt(0);
#else
  asm volatile("s_wait_asynccnt 0x0" ::: "memory");
#endif
}

// ---------------- small math helpers ----------------
__device__ __forceinline__ void inv3(const float m[9], float o[9]) {
  float a=m[0],b=m[1],c=m[2],d=m[3],e=m[4],f=m[5],g=m[6],h=m[7],i=m[8];
  float A  = e*i - f*h;
  float Bc = f*g - d*i;
  float Cc = d*h - e*g;
  float r  = 1.0f / (a*A + b*Bc + c*Cc);
  o[0]=A*r;   o[1]=(c*h-b*i)*r; o[2]=(b*f-c*e)*r;
  o[3]=Bc*r;  o[4]=(a*i-c*g)*r; o[5]=(c*d-a*f)*r;
  o[6]=Cc*r;  o[7]=(b*g-a*h)*r; o[8]=(a*e-b*d)*r;
}

__device__ __forceinline__ float wave_max(float v) {
  for (int off = 16; off > 0; off >>= 1) v = fmaxf(v, __shfl_xor(v, off, 32));
  return v;
}
__device__ __forceinline__ float wave_sum(float v) {
  for (int off = 16; off > 0; off >>= 1) v += __shfl_xor(v, off, 32);
  return v;
}

// frustum point (xs,ys,depth) -> ego -> voxel key (ix*200+iy) or -1 if dropped.
// NOTE: (int) cast truncates toward zero, matching reference astype(int32).
__device__ __forceinline__ int vox_key(float depth, float xs, float ys,
                                       const float* Pinv, const float* t1,
                                       const float* cmb,  const float* t2) {
  float ux = xs - t1[0], uy = ys - t1[1], uz = depth - t1[2];
  float px = Pinv[0]*ux + Pinv[1]*uy + Pinv[2]*uz;
  float py = Pinv[3]*ux + Pinv[4]*uy + Pinv[5]*uz;
  float pz = Pinv[6]*ux + Pinv[7]*uy + Pinv[8]*uz;
  float qx = px * pz, qy = py * pz, qz = pz;       // unproject
  float gx = cmb[0]*qx + cmb[1]*qy + cmb[2]*qz + t2[0];
  float gy = cmb[3]*qx + cmb[4]*qy + cmb[5]*qz + t2[1];
  float gz = cmb[6]*qx + cmb[7]*qy + cmb[8]*qz + t2[2];
  int ix = (int)((gx + 50.0f) / 0.5f);
  int iy = (int)((gy + 50.0f) / 0.5f);
  int iz = (int)((gz + 10.0f) / 20.0f);
  bool kept = (ix >= 0) & (ix < NXX) & (iy >= 0) & (iy < NXY) & (iz == 0);
  return kept ? (ix * NXY + iy) : -1;
}

// ---------------- splat body (one wave == one pixel) ----------------
// Grid: x = w-tile (6 tiles of 8 waves), y = h (16), z = bn (48).
// FAST: accumulate into (B,X,Y,C) scratch -> two 128B-coalesced atomic ops per
// (pixel, depth-bin). !FAST: straight into (B,C,X,Y) output (scattered).
__device__ __forceinline__ void lss_splat_body(
    bool FAST,
    const float* __restrict__ feat,   const float* __restrict__ rots,
    const float* __restrict__ trans,  const float* __restrict__ intrins,
    const float* __restrict__ prots,  const float* __restrict__ ptrans,
    float* __restrict__ dst) {
  const int lane = threadIdx.x & 31;
  const int warp = threadIdx.x >> 5;
  const int w    = blockIdx.x * 8 + warp;   // 0..47 (44..47 idle)
  const int h    = blockIdx.y;              // 0..15
  const int bn   = blockIdx.z;              // 0..47
  if (w >= LSS_FW) return;                  // uniform per wave; no barriers used
  const int b  = bn / LSS_N;
  const int pp = h * LSS_FW + w;

  // Stage this pixel's 105 channel values (41 depth logits + 64 features)
  // into LDS with 4 async ops (channel = lane + 32k, stride FH*FW floats).
  __shared__ float stage[8][112];
  float* sw = stage[warp];
  const float* g = feat + (size_t)bn * LSS_CH * LSS_PP + pp;
  async_ld32(g + (size_t)(lane +  0) * LSS_PP, sw + lane +  0);
  async_ld32(g + (size_t)(lane + 32) * LSS_PP, sw + lane + 32);
  async_ld32(g + (size_t)(lane + 64) * LSS_PP, sw + lane + 64);
  if (lane < LSS_CH - 96)
    async_ld32(g + (size_t)(lane + 96) * LSS_PP, sw + lane + 96);

  // Per-(b,n) uniform matrices, computed while the async loads are in flight.
  float R[9], K[9], P[9], Pinv[9], Kinv[9], cmb[9], t1[3], t2[3];
  for (int i = 0; i < 9; ++i) {
    R[i] = rots[bn * 9 + i]; K[i] = intrins[bn * 9 + i]; P[i] = prots[bn * 9 + i];
  }
  for (int i = 0; i < 3; ++i) { t1[i] = ptrans[bn * 3 + i]; t2[i] = trans[bn * 3 + i]; }
  inv3(P, Pinv);
  inv3(K, Kinv);
  for (int r = 0; r < 3; ++r)
    for (int c = 0; c < 3; ++c)
      cmb[r*3+c] = R[r*3+0]*Kinv[0*3+c] + R[r*3+1]*Kinv[1*3+c] + R[r*3+2]*Kinv[2*3+c];

  async_wait0();
  volatile const float* sv = stage[warp];
  const float l0 = sv[lane];                                   // logit d=lane
  const float l1 = (lane < 9) ? sv[lane + 32] : -INFINITY;     // logit d=lane+32
  const float f0 = sv[LSS_D + lane];                           // feature c=lane
  const float f1 = sv[LSS_D + lane + 32];                      // feature c=lane+32

  // softmax over D=41 across the wave
  const float m  = wave_max(fmaxf(l0, l1));
  const float e0 = expf(l0 - m);
  const float e1 = (lane < 9) ? expf(l1 - m) : 0.0f;
  const float s  = wave_sum(e0 + e1);
  const float w0 = e0 / s;
  const float w1 = e1 / s;

  // geometry for this lane's depth bins (linspace: xs=w*703/43, ys=h*17, ds=4+d)
  const float xs = (float)w * (703.0f / 43.0f);
  const float ys = (float)h * 17.0f;
  const int k0 = vox_key(4.0f  + (float)lane, xs, ys, Pinv, t1, cmb, t2);
  const int k1 = (lane < 9) ? vox_key(36.0f + (float)lane, xs, ys, Pinv, t1, cmb, t2) : -1;

  // scatter: broadcast (key, weight) per depth bin with v_readlane -> uniform branch
#pragma unroll
  for (int d = 0; d < LSS_D; ++d) {
    int kd; float wd;
    if (d < 32) {
      kd = __builtin_amdgcn_readlane(k0, d);
      wd = __int_as_float(__builtin_amdgcn_readlane(__float_as_int(w0), d));
    } else {
      kd = __builtin_amdgcn_readlane(k1, d - 32);
      wd = __int_as_float(__builtin_amdgcn_readlane(__float_as_int(w1), d - 32));
    }
    if (kd >= 0) {
      if (FAST) {
        // (B,X,Y,C): 64 channels contiguous -> two 128B-coalesced atomic ops
        float* p = dst + ((size_t)b * BEV_CELLS + (size_t)kd) * LSS_C;
        unsafeAtomicAdd(p + lane,      wd * f0);
        unsafeAtomicAdd(p + lane + 32, wd * f1);
      } else {
        // (B,C,X,Y) directly
        float* p = dst + (size_t)b * ((size_t)LSS_C * BEV_CELLS) + (size_t)kd;
        unsafeAtomicAdd(p + (size_t)lane * BEV_CELLS,        wd * f0);
        unsafeAtomicAdd(p + (size_t)(lane + 32) * BEV_CELLS, wd * f1);
      }
    }
  }
}

// Defined first so the disasm snippet shows the hot kernel.
__global__ __launch_bounds__(256) void lss_splat_fast(
    const float* __restrict__ feat,   const float* __restrict__ rots,
    const float* __restrict__ trans,  const float* __restrict__ intrins,
    const float* __restrict__ prots,  const float* __restrict__ ptrans,
    float* __restrict__ dst) {
  lss_splat_body(true, feat, rots, trans, intrins, prots, ptrans, dst);
}

__global__ __launch_bounds__(256) void lss_splat_slow(
    const float* __restrict__ feat,   const float* __restrict__ rots,
    const float* __restrict__ trans,  const float* __restrict__ intrins,
    const float* __restrict__ prots,  const float* __restrict__ ptrans,
    float* __restrict__ dst) {
  lss_splat_body(false, feat, rots, trans, intrins, prots, ptrans, dst);
}

// ---------------- aux kernels ----------------
__global__ __launch_bounds__(256) void zero_f4(float4* p, int n4) {
  int i = blockIdx.x * blockDim.x + threadIdx.x;
  if (i < n4) p[i] = make_float4(0.f, 0.f, 0.f, 0.f);
}

// (B,X,Y,C) -> (B,C,X,Y) tiled transpose, conflict-free LDS, coalesced both ways.
__global__ __launch_bounds__(256) void bev_transpose(const float* __restrict__ src,
                                                     float* __restrict__ dst) {
  __shared__ float tile[32][65];
  const int bt = blockIdx.x;
  const int yt = bt % 7;
  const int x  = (bt / 7) % NXX;
  const int b  = bt / (7 * NXX);
  const int y0 = yt * 32;
#pragma unroll
  for (int k = 0; k < 8; ++k) {
    int li = threadIdx.x + k * 256;
    int c = li & 63, yy = li >> 6;
    int y = y0 + yy;
    if (y < NXY) tile[yy][c] = src[((size_t)(b * NXX + x) * NXY + y) * LSS_C + c];
  }
  __syncthreads();
#pragma unroll
  for (int k = 0; k < 8; ++k) {
    int li = threadIdx.x + k * 256;
    int yy = li & 31, c = li >> 5;
    int y = y0 + yy;
    if (y < NXY) dst[((size_t)(b * LSS_C + c) * NXX + x) * NXY + y] = tile[yy][c];
  }
}

// ---------------- host ----------------
extern "C" void kernel_launch(void* const* d_in, const int* in_sizes, int n_in,
                              void* d_out, int out_size, void* d_ws, size_t ws_size,
                              hipStream_t stream) {
  (void)in_sizes; (void)n_in; (void)out_size;
  const float* feat    = (const float*)d_in[0];
  const float* rots    = (const float*)d_in[1];
  const float* trans   = (const float*)d_in[2];
  const float* intrins = (const float*)d_in[3];
  const float* prots   = (const float*)d_in[4];
  const float* ptrans  = (const float*)d_in[5];
  float* out = (float*)d_out;

  const int n4 = (int)(BEV_ELEMS / 4);               // 5,120,000 float4
  const int zb = (n4 + 255) / 256;
  const dim3 splat_grid(6, LSS_FH, LSS_B * LSS_N);   // w-tiles, h, bn

  if (ws_size >= BEV_ELEMS * sizeof(float)) {
    float* ws = (float*)d_ws;
    zero_f4<<<zb, 256, 0, stream>>>((float4*)ws, n4);
    lss_splat_fast<<<splat_grid, 256, 0, stream>>>(feat, rots, trans, intrins,
                                                   prots, ptrans, ws);
    bev_transpose<<<LSS_B * NXX * 7, 256, 0, stream>>>(ws, out);
  } else {
    zero_f4<<<zb, 256, 0, stream>>>((float4*)out, n4);
    lss_splat_slow<<<splat_grid, 256, 0, stream>>>(feat, rots, trans, intrins,
                                                   prots, ptrans, out);
  }
}